// SAGE_78348793413775
// MI455X (gfx1250) — compile-verified
//
#include <hip/hip_runtime.h>

// ---------------------------------------------------------------------------
// GraphSAGE 2-layer forward for MI455X (gfx1250, wave32).
//   layer1: h   = relu( mean_agg(x)  @ W1l + b1 + x @ W1r )   [50000 x 128]
//   layer2: out =       mean_agg(h)  @ W2l + b2 + h @ W2r     [50000 x 64]
// Edge scatter is the memory-bound part (~0.9 GB of traffic); GEMMs are tiny
// (4.9 GFLOP) so we keep full fp32 precision and use V_WMMA_F32_16X16X4_F32.
// ---------------------------------------------------------------------------

#define N_NODES 50000
#define N_EDGES 800000
#define IN_C    128
#define HID_C   128
#define OUT_C   64

typedef float v2f __attribute__((ext_vector_type(2)));
typedef float v8f __attribute__((ext_vector_type(8)));

// ---------------------------------------------------------------------------
// Zero a float buffer (count must be a multiple of 4).
// ---------------------------------------------------------------------------
__global__ void sage_zero_kernel(float* __restrict__ p, long long n4) {
    long long i = (long long)blockIdx.x * blockDim.x + threadIdx.x;
    if (i < n4) ((float4*)p)[i] = make_float4(0.f, 0.f, 0.f, 0.f);
}

// ---------------------------------------------------------------------------
// Scatter-add of 128-wide feature rows along edges (one wave32 per edge,
// 4 channels per lane) + degree accumulation (first layer only).
// ---------------------------------------------------------------------------
__global__ __launch_bounds__(256) void sage_scatter_kernel(
    const float* __restrict__ feat,
    const int*   __restrict__ src,
    const int*   __restrict__ dst,
    float* __restrict__ sum,
    float* __restrict__ deg,      // nullptr -> skip degree accumulation
    int n_edges) {
    const int lane = threadIdx.x & 31;
    const int e    = blockIdx.x * (blockDim.x >> 5) + (threadIdx.x >> 5);
    if (e >= n_edges) return;
    const int s = src[e];
    const int d = dst[e];
    const float4 v = ((const float4*)(feat + (long long)s * 128))[lane];
    float* o = sum + (long long)d * 128 + lane * 4;
    atomicAdd(o + 0, v.x);
    atomicAdd(o + 1, v.y);
    atomicAdd(o + 2, v.z);
    atomicAdd(o + 3, v.w);
    if (deg != nullptr && lane == 0) atomicAdd(deg + d, 1.0f);
}

// ---------------------------------------------------------------------------
// Fused SAGE linear:  out[16xCO tile] = (sum/deg') @ Wl + bias + feat @ Wr
// One wave32 per 16-node tile; weights staged in LDS; fp32 WMMA 16x16x4.
//   A fragment (16x4 f32): lanes 0-15 rows M=0..15 K=4k+{0,1};
//                          lanes 16-31 rows M=0..15 K=4k+{2,3}.
//   B fragment (4x16 f32): mirrored over N.
//   C/D (16x16 f32): VGPR v -> M = v + 8*(lane>=16), N = lane&15.
// ---------------------------------------------------------------------------
template <int CO, bool RELU>
__global__ __launch_bounds__(256) void sage_gemm_kernel(
    const float* __restrict__ sum,
    const float* __restrict__ deg,
    const float* __restrict__ feat,
    const float* __restrict__ Wl,     // [128 x CO] row-major
    const float* __restrict__ bias,   // [CO]
    const float* __restrict__ Wr,     // [128 x CO] row-major
    float* __restrict__ out,          // [N x CO]
    int n_tiles) {
    constexpr int NT = CO / 16;

    extern __shared__ float smem[];
    float* sWl = smem;                // 128*CO floats
    float* sWr = smem + 128 * CO;     // 128*CO floats

    const int tid = threadIdx.x;
    for (int i = tid; i < 128 * CO; i += blockDim.x) {
        sWl[i] = Wl[i];
        sWr[i] = Wr[i];
    }
    __syncthreads();

    const int wave = tid >> 5;
    const int lane = tid & 31;
    const int tile = blockIdx.x * (blockDim.x >> 5) + wave;
    if (tile >= n_tiles) return;      // whole wave exits together (EXEC all-1s for WMMA)

    const int row0 = tile * 16;
    const int hi   = lane >> 4;       // 0: K lo pair / M 0..7 half, 1: K hi pair / M 8..15 half
    const int lr   = lane & 15;       // A row within tile; B/C column within N tile
    const int arow = row0 + lr;

    const float dg   = deg[arow];
    const float invd = 1.0f / fmaxf(dg, 1.0f);

    const float* aggRow = sum  + (long long)arow * 128;
    const float* xRow   = feat + (long long)arow * 128;

    v8f acc[NT];
#pragma unroll
    for (int n = 0; n < NT; ++n) acc[n] = (v8f){};

    for (int k = 0; k < 32; ++k) {            // K = 128 in steps of 4
        const int c = 4 * k + 2 * hi;         // 8-byte aligned float2 per lane
        v2f aAgg, aX;
        aAgg.x = aggRow[c]     * invd;
        aAgg.y = aggRow[c + 1] * invd;
        aX.x   = xRow[c];
        aX.y   = xRow[c + 1];
#pragma unroll
        for (int n = 0; n < NT; ++n) {
            const int col = n * 16 + lr;
            v2f bL, bR;
            bL.x = sWl[(c)     * CO + col];
            bL.y = sWl[(c + 1) * CO + col];
            bR.x = sWr[(c)     * CO + col];
            bR.y = sWr[(c + 1) * CO + col];
            acc[n] = __builtin_amdgcn_wmma_f32_16x16x4_f32(
                false, aAgg, false, bL, (short)0, acc[n], false, false);
            acc[n] = __builtin_amdgcn_wmma_f32_16x16x4_f32(
                false, aX,   false, bR, (short)0, acc[n], false, false);
        }
    }

    // Epilogue: bias add (+ ReLU), store per documented C/D layout.
#pragma unroll
    for (int n = 0; n < NT; ++n) {
        const int col = n * 16 + lr;
        const float b = bias[col];
#pragma unroll
        for (int v = 0; v < 8; ++v) {
            const int r = row0 + v + 8 * hi;
            float val = acc[n][v] + b;
            if (RELU) val = fmaxf(val, 0.0f);
            out[(long long)r * CO + col] = val;
        }
    }
}

// ---------------------------------------------------------------------------
// Host-side orchestration (all on `stream`, graph-capture safe).
// Workspace layout (floats): [sum: N*128][deg: N][h: N*128]  (~51.4 MB)
// ---------------------------------------------------------------------------
extern "C" void kernel_launch(void* const* d_in, const int* in_sizes, int n_in,
                              void* d_out, int out_size, void* d_ws, size_t ws_size,
                              hipStream_t stream) {
    const float* x   = (const float*)d_in[0];
    const int*   ei  = (const int*)  d_in[1];   // [2, E]: src row then dst row
    const float* W1l = (const float*)d_in[2];
    const float* b1  = (const float*)d_in[3];
    const float* W1r = (const float*)d_in[4];
    const float* W2l = (const float*)d_in[5];
    const float* b2  = (const float*)d_in[6];
    const float* W2r = (const float*)d_in[7];

    const int* src = ei;
    const int* dst = ei + N_EDGES;

    float* ws  = (float*)d_ws;
    float* sum = ws;                                   // N*128 floats
    float* deg = ws + (long long)N_NODES * 128;        // N floats
    float* h   = deg + N_NODES;                        // N*128 floats
    float* out = (float*)d_out;                        // N*64 floats

    const int tiles       = N_NODES / 16;              // 3125 (exact)
    const int gemm_blocks = (tiles + 7) / 8;           // 8 waves/block
    const int edge_blocks = (N_EDGES + 7) / 8;         // 8 edges/block (1 wave each)

    // ---- Layer 1 -----------------------------------------------------------
    {
        const long long n = (long long)N_NODES * 128 + N_NODES;  // sum + deg (contig)
        const long long n4 = n / 4;
        sage_zero_kernel<<<(int)((n4 + 255) / 256), 256, 0, stream>>>(sum, n4);
    }
    sage_scatter_kernel<<<edge_blocks, 256, 0, stream>>>(x, src, dst, sum, deg, N_EDGES);
    sage_gemm_kernel<HID_C, true><<<gemm_blocks, 256, 2 * 128 * HID_C * sizeof(float), stream>>>(
        sum, deg, x, W1l, b1, W1r, h, tiles);

    // ---- Layer 2 (deg reused; only the running sum is re-zeroed) -----------
    {
        const long long n  = (long long)N_NODES * 128;
        const long long n4 = n / 4;
        sage_zero_kernel<<<(int)((n4 + 255) / 256), 256, 0, stream>>>(sum, n4);
    }
    sage_scatter_kernel<<<edge_blocks, 256, 0, stream>>>(h, src, dst, sum, nullptr, N_EDGES);
    sage_gemm_kernel<OUT_C, false><<<gemm_blocks, 256, 2 * 128 * OUT_C * sizeof(float), stream>>>(
        sum, deg, h, W2l, b2, W2r, out, tiles);
}